// Slot_Dec_60885456388756
// MI455X (gfx1250) — compile-verified
//
#include <hip/hip_runtime.h>
#include <hip/hip_bf16.h>
#include <math.h>

// ---------------------------------------------------------------------------
// LSTM-with-batch-scan for MI455X (gfx1250).
//   Phase A: fp32 WMMA GEMM (V_WMMA_F32_16X16X4_F32) for all input-dependent
//            pre-gates, + global_prefetch for the next K tile.
//   Phase B: 16-WGP cooperative serial scan; W_hh slice resident in 320KB LDS;
//            TDM tensor_load_to_lds stages the strided out_prev tile,
//            global_load_async_to_lds_b32 stages the h vector each chain step;
//            per-step cross-WGP sync (cluster barrier fast path + atomic bar).
// ---------------------------------------------------------------------------

typedef float v2f __attribute__((ext_vector_type(2)));
typedef float v8f __attribute__((ext_vector_type(8)));
typedef unsigned int u32x4 __attribute__((ext_vector_type(4)));
typedef int i32x4 __attribute__((ext_vector_type(4)));
typedef int i32x8 __attribute__((ext_vector_type(8)));

namespace {
constexpr int kH   = 512;           // hidden
constexpr int kB   = 64;            // batch
constexpr int kS   = 256;           // seqlen
constexpr int kG   = 4 * kH;        // 2048 gates
constexpr int k5H  = 5 * kH;        // 2560 (W_ih row stride)
constexpr int kNWG = 16;            // scan workgroups (gate slices, g % 16 == wg)
constexpr int kSlice = kG / kNWG;   // 128 gates per WG
constexpr int KSTEP = 16;           // GEMM K tile

// d_ws layout
constexpr size_t kBarOff  = 0;      // 2 x u32 barrier state
constexpr size_t kHbufOff = 256;    // 2 x 512 f32 ping-pong h
constexpr size_t kGxOff   = 8192;   // 16384 x 2048 f32 pre-gates (permuted cols)

// Phase-B dynamic LDS layout (floats)
constexpr int kWldsStride = kH + 1;                 // 513, bank-conflict pad
constexpr int kWldsF  = kSlice * kWldsStride;       // 65664
constexpr int kHlF    = kH;                         // 512
constexpr int kPartF  = 256;
constexpr int kRbufF  = kSlice;                     // 128
constexpr int kCldsF  = 32;
constexpr int kPbufF  = kB * kSlice;                // 8192
constexpr int kOpcStride = 65;                      // 64 + 1 pad DWORD per 64
constexpr int kOpcF   = kB * kOpcStride;            // 4160
constexpr int kSmemFloats = kWldsF + kHlF + kPartF + kRbufF + kCldsF + kPbufF + kOpcF;
constexpr size_t kSmemBytes = (size_t)kSmemFloats * 4;  // 315,776 B < 320KB WGP LDS
}

// ---------- LDS byte-offset of a generic pointer (AS3 addrspacecast) --------
__device__ __forceinline__ unsigned lds_offset_of(const void* p) {
  return (unsigned)(unsigned long long)(__attribute__((address_space(3))) const void*)p;
}

// ---------- TDM: 2D tile (data_size = 4B) global -> LDS ----------------------
// D# per CDNA5 ISA §8.3/8.4. tile_d0 elements per row, tile_d1 rows,
// row_stride in elements; optional +1-DWORD LDS pad every (2<<pad_code) DWORDs.
#if __has_builtin(__builtin_amdgcn_tensor_load_to_lds)
#define HAVE_TDM 1
__device__ __forceinline__ void tdm_load_2d_f32(unsigned lds_addr, const void* gaddr,
                                                unsigned tile_d0, unsigned tile_d1,
                                                unsigned long long row_stride_elems,
                                                unsigned pad_code, bool pad_en) {
  unsigned long long ga = (unsigned long long)gaddr;
  u32x4 g0;
  g0[0] = 1u;                                                   // count=1 (valid)
  g0[1] = lds_addr;                                             // lds_addr
  g0[2] = (unsigned)ga;                                         // global_addr lo
  g0[3] = (unsigned)((ga >> 32) & 0x1FFFFFFull) | (2u << 30);   // addr hi | type=2
  i32x8 g1;
  g1[0] = (int)((2u << 16) |                                    // data_size = 4B
                (pad_en ? (1u << 20) : 0u) |                    // pad_enable
                ((pad_code & 7u) << 22));                       // pad_interval
  g1[1] = (int)((tile_d0 & 0xFFFFu) << 16);                     // tensor_dim0 lo16
  g1[2] = (int)((tile_d0 >> 16) | ((tile_d1 & 0xFFFFu) << 16)); // dim0 hi | dim1 lo
  g1[3] = (int)((tile_d1 >> 16) | (tile_d0 << 16));             // dim1 hi | tile_dim0
  g1[4] = (int)(tile_d1 & 0xFFFFu);                             // tile_dim1 (dim2=0)
  g1[5] = (int)(row_stride_elems & 0xFFFFFFFFull);              // dim0_stride lo32
  g1[6] = (int)((row_stride_elems >> 32) & 0xFFFFull);          // dim0_stride hi16
  g1[7] = 0;
  i32x4 z4 = {0, 0, 0, 0};
#if defined(__clang_major__) && (__clang_major__ >= 23)
  i32x8 z8 = {0, 0, 0, 0, 0, 0, 0, 0};
  __builtin_amdgcn_tensor_load_to_lds(g0, g1, z4, z4, z8, 0);
#else
  __builtin_amdgcn_tensor_load_to_lds(g0, g1, z4, z4, 0);
#endif
}
#else
#define HAVE_TDM 0
#endif

// ---------- async global -> LDS, one DWORD per lane --------------------------
__device__ __forceinline__ void async_g2l_b32(unsigned lds_byte_addr, const void* gp) {
  unsigned long long ga = (unsigned long long)gp;
  asm volatile("global_load_async_to_lds_b32 %0, %1, off"
               :: "v"(lds_byte_addr), "v"(ga) : "memory");
}
__device__ __forceinline__ void wait_asynccnt0() {
  asm volatile("s_wait_asynccnt 0" ::: "memory");
}

// ---------------------------------------------------------------------------
// Phase A: Gx[t, perm(n)] = XC[t,:] @ W_ih[n, 0:2048]^T + (b_ih[n]+b_hh[n])
//   XC[t] = concat(x[b,s,:], hi[b,s,:]),  t = s*64 + b
//   perm(n) = (n%16)*128 + n/16  -> each scan WG reads a contiguous 128-slice.
// Block tile 128(M) x 64(N), 8 waves (4x2), wave tile 32x32 = 2x2 WMMA tiles,
// fp32 V_WMMA_F32_16X16X4_F32, K streamed through LDS in steps of 16.
// ---------------------------------------------------------------------------
__global__ __launch_bounds__(256)
void pregate_gemm(const float* __restrict__ x, const float* __restrict__ hi,
                  const float* __restrict__ Wih, const float* __restrict__ bih,
                  const float* __restrict__ bhh, float* __restrict__ Gx) {
  __shared__ float As[128][KSTEP + 1];
  __shared__ float Bs[64][KSTEP + 1];

  const int tid  = threadIdx.x;
  const int lane = tid & 31;
  const int wid  = tid >> 5;
  const int wm   = wid & 3;          // wave M position (0..3)
  const int wn   = wid >> 2;         // wave N position (0..1)
  const int blkM = blockIdx.x & 127; // 16384/128 = 128 M blocks
  const int blkN = blockIdx.x >> 7;  // 2048/64   = 32  N blocks
  const int m0   = blkM * 128;
  const int n0   = blkN * 64;
  const int lan      = lane & 15;
  const int laneHalf = lane >> 4;

  v8f acc[2][2];
  for (int mi = 0; mi < 2; ++mi)
    for (int ni = 0; ni < 2; ++ni)
      for (int q = 0; q < 8; ++q) acc[mi][ni][q] = 0.0f;

  for (int k0 = 0; k0 < 2048; k0 += KSTEP) {
    // Prefetch next K tile into the cache hierarchy (global_prefetch_b8).
    if (k0 + KSTEP < 2048) {
      const int kn = k0 + KSTEP;
      if (tid < 128) {
        int m = m0 + tid, b = m & 63, si = m >> 6;
        const float* ap = (kn < 1024)
            ? &x[((size_t)b * kS + si) * 1024 + kn]
            : &hi[((size_t)b * kS + si) * 1024 + (kn - 1024)];
        __builtin_prefetch(ap, 0, 0);
      } else {
        __builtin_prefetch(&Wih[(size_t)(n0 + (tid & 63)) * k5H + kn], 0, 0);
      }
    }
    __syncthreads();
    // A tile: 128 rows x 16 k  (gathered: row m -> (b = m&63, s = m>>6))
#pragma unroll
    for (int i = 0; i < 8; ++i) {
      int e = tid + i * 256;
      int r = e >> 4, k = e & 15;
      int m = m0 + r;
      int b = m & 63, s = m >> 6;
      int kk = k0 + k;
      float v;
      if (kk < 1024) v = x[((size_t)b * kS + s) * 1024 + kk];
      else           v = hi[((size_t)b * kS + s) * 1024 + (kk - 1024)];
      As[r][k] = v;
    }
    // B tile: 64 W_ih rows x 16 k (row stride 2560)
#pragma unroll
    for (int i = 0; i < 4; ++i) {
      int e = tid + i * 256;
      int n = e >> 4, k = e & 15;
      Bs[n][k] = Wih[(size_t)(n0 + n) * k5H + k0 + k];
    }
    __syncthreads();

#pragma unroll
    for (int kk = 0; kk < KSTEP; kk += 4) {
      const int kf = kk + laneHalf * 2;  // lanes 0-15: k,k+1 ; lanes 16-31: k+2,k+3
      v2f afrag[2], bfrag[2];
#pragma unroll
      for (int mi = 0; mi < 2; ++mi) {
        int row = wm * 32 + mi * 16 + lan;
        afrag[mi].x = As[row][kf];
        afrag[mi].y = As[row][kf + 1];
      }
#pragma unroll
      for (int ni = 0; ni < 2; ++ni) {
        int col = wn * 32 + ni * 16 + lan;
        bfrag[ni].x = Bs[col][kf];
        bfrag[ni].y = Bs[col][kf + 1];
      }
#pragma unroll
      for (int mi = 0; mi < 2; ++mi)
#pragma unroll
        for (int ni = 0; ni < 2; ++ni)
          acc[mi][ni] = __builtin_amdgcn_wmma_f32_16x16x4_f32(
              false, afrag[mi], false, bfrag[ni], (short)0, acc[mi][ni],
              false, false);
    }
  }

  // Epilogue: +bias, permuted-column store.
#pragma unroll
  for (int mi = 0; mi < 2; ++mi) {
#pragma unroll
    for (int ni = 0; ni < 2; ++ni) {
      const int n = n0 + wn * 32 + ni * 16 + lan;
      const float bias = bih[n] + bhh[n];
      const size_t ncol = (size_t)((n & 15) * kSlice + (n >> 4));
#pragma unroll
      for (int r = 0; r < 8; ++r) {
        const int m = m0 + wm * 32 + mi * 16 + laneHalf * 8 + r;
        Gx[(size_t)m * kG + ncol] = acc[mi][ni][r] + bias;
      }
    }
  }
}

// ---------------------------------------------------------------------------
// Cross-WGP barrier: cluster barrier (fast path / arch NOP) + sense-reversing
// device-scope atomic barrier in d_ws. cnt=bar[0], gen=bar[1].
// ---------------------------------------------------------------------------
__device__ __forceinline__ void grid_barrier(unsigned* bar, int tid) {
  __threadfence();            // each wave publishes its global stores
  __syncthreads();
  if (tid == 0) {
    __builtin_amdgcn_s_cluster_barrier();  // s_barrier_signal/wait -3 (NOP if unclustered)
    unsigned* cnt = bar;
    unsigned* gen = bar + 1;
    unsigned g = __hip_atomic_load(gen, __ATOMIC_RELAXED, __HIP_MEMORY_SCOPE_AGENT);
    unsigned prev = __hip_atomic_fetch_add(cnt, 1u, __ATOMIC_ACQ_REL,
                                           __HIP_MEMORY_SCOPE_AGENT);
    if (prev == (unsigned)(kNWG - 1)) {
      __hip_atomic_store(cnt, 0u, __ATOMIC_RELAXED, __HIP_MEMORY_SCOPE_AGENT);
      __hip_atomic_store(gen, g + 1u, __ATOMIC_RELEASE, __HIP_MEMORY_SCOPE_AGENT);
    } else {
      while (__hip_atomic_load(gen, __ATOMIC_ACQUIRE, __HIP_MEMORY_SCOPE_AGENT) == g)
        __builtin_amdgcn_s_sleep(2);
    }
  }
  __syncthreads();
  __threadfence();            // acquire: drop stale lines before re-reading h
}

// ---------------------------------------------------------------------------
// Phase B: 16 WGs x 256 threads, persistent. WG j owns gates {g : g%16==j}
// (local l = g/16 in 0..127) and h/c elements {e : e%16==j} -> i/f/g/o of its
// h elements are all locally owned. Per chain step t:
//   gates = Gx[t] + Pbuf[b] + Wlds @ h_{t-1};  update h,c;  publish 32 h;
//   one grid barrier. Pbuf (out_prev @ W_ih[:,4H:]^T) rebuilt once per
//   timestep from d_out via TDM tensor tiles.
// ---------------------------------------------------------------------------
__global__ __launch_bounds__(256)
void lstm_scan(const float* __restrict__ Whh, const float* __restrict__ Wih,
               const float* __restrict__ Gx, float* __restrict__ out,
               float* __restrict__ hbuf, unsigned* __restrict__ bar) {
  extern __shared__ char smem_dyn[];
  float* Wlds = (float*)smem_dyn;         // [128][513]
  float* hl   = Wlds + kWldsF;            // [512]
  float* part = hl + kHlF;                // [256]
  float* rbuf = part + kPartF;            // [128]
  float* clds = rbuf + kRbufF;            // [32]
  float* Pbuf = clds + kCldsF;            // [64][128]
  float* opc  = Pbuf + kPbufF;            // [64][65]

  const int j   = blockIdx.x;             // slice id 0..15
  const int tid = threadIdx.x;

  // Stage this WG's W_hh slice into LDS (stays resident for all 16384 steps).
  for (int e = tid; e < kSlice * kH; e += 256) {
    int l = e >> 9, k = e & (kH - 1);
    Wlds[l * kWldsStride + k] = Whh[(size_t)(l * 16 + j) * kH + k];
  }
  if (tid < 32) clds[tid] = 0.0f;
  __syncthreads();

  for (int s = 0; s < kS; ++s) {
    // ---------------- out_prev GEMM slice -> Pbuf ----------------
    if (s == 0) {
      for (int e = tid; e < kPbufF; e += 256) Pbuf[e] = 0.0f;
      __syncthreads();
    } else {
      const int l  = tid & 127;
      const int bg = tid >> 7;                       // thread owns b = 2c+bg
      const int g  = l * 16 + j;
      const float* wp = Wih + (size_t)g * k5H + 2048;
      float acc[32];
#pragma unroll
      for (int c = 0; c < 32; ++c) acc[c] = 0.0f;
      for (int kc = 0; kc < kH / 64; ++kc) {
        const int kk0 = kc * 64;
        __syncthreads();                             // opc reuse guard
#if HAVE_TDM
        // TDM: 64x64 f32 tile of out[:, s-1, kk0:kk0+64], row stride S*H,
        // +1 DWORD LDS pad per 64 DWORDs -> opc stride 65.
        if (tid == 0) {
          tdm_load_2d_f32(lds_offset_of(opc),
                          out + (size_t)(s - 1) * kH + kk0,
                          /*tile_d0=*/64u, /*tile_d1=*/(unsigned)kB,
                          /*row_stride=*/(unsigned long long)kS * kH,
                          /*pad_code(64 DW)=*/5u, /*pad_en=*/true);
          __builtin_amdgcn_s_wait_tensorcnt(0);
        }
#else
        for (int e = tid; e < kB * 64; e += 256) {
          int b = e >> 6, kk = e & 63;
          opc[b * kOpcStride + kk] =
              out[(size_t)b * kS * kH + (size_t)(s - 1) * kH + kk0 + kk];
        }
#endif
        __syncthreads();
        for (int kk = 0; kk < 64; ++kk) {
          const float w = wp[kk0 + kk];
#pragma unroll
          for (int c = 0; c < 32; ++c)
            acc[c] = fmaf(opc[(c * 2 + bg) * kOpcStride + kk], w, acc[c]);
        }
      }
      __syncthreads();
#pragma unroll
      for (int c = 0; c < 32; ++c) Pbuf[(c * 2 + bg) * kSlice + l] = acc[c];
      __syncthreads();
    }

    // ---------------- serial chain over batch ----------------
    for (int b = 0; b < kB; ++b) {
      const int t = s * kB + b;

      // h_{t-1} -> LDS (async global->LDS, ASYNCcnt-tracked)
      if (t == 0) {
        for (int k = tid; k < kH; k += 256) hl[k] = 0.0f;
      } else {
        const float* hp = hbuf + (size_t)((t - 1) & 1) * kH;
        for (int k = tid; k < kH; k += 256)
          async_g2l_b32(lds_offset_of(hl + k), hp + k);
        wait_asynccnt0();
      }
      __syncthreads();

      // 128x512 matvec: thread = (l = tid&127, k-half = tid>>7)
      {
        const int l  = tid & 127;
        const int kh = tid >> 7;
        const float* wrow = Wlds + l * kWldsStride + kh * 256;
        const float* hv   = hl + kh * 256;
        float a0 = 0.f, a1 = 0.f, a2 = 0.f, a3 = 0.f;
#pragma unroll 4
        for (int k = 0; k < 256; k += 4) {
          a0 = fmaf(wrow[k + 0], hv[k + 0], a0);
          a1 = fmaf(wrow[k + 1], hv[k + 1], a1);
          a2 = fmaf(wrow[k + 2], hv[k + 2], a2);
          a3 = fmaf(wrow[k + 3], hv[k + 3], a3);
        }
        part[tid] = (a0 + a1) + (a2 + a3);
      }
      __syncthreads();

      if (tid < kSlice) {
        rbuf[tid] = part[tid] + part[tid + 128]
                  + Gx[(size_t)t * kG + (size_t)j * kSlice + tid]
                  + Pbuf[b * kSlice + tid];
      }
      __syncthreads();

      if (tid < 32) {
        const float ig = 1.0f / (1.0f + __expf(-rbuf[tid]));
        const float fg = 1.0f / (1.0f + __expf(-rbuf[32 + tid]));
        const float gg = tanhf(rbuf[64 + tid]);
        const float og = 1.0f / (1.0f + __expf(-rbuf[96 + tid]));
        const float c  = fg * clds[tid] + ig * gg;
        clds[tid] = c;
        const float h = og * tanhf(c);
        const int e = tid * 16 + j;
        hbuf[(size_t)(t & 1) * kH + e] = h;
        out[(size_t)b * kS * kH + (size_t)s * kH + e] = h;
      }

      grid_barrier(bar, tid);
    }
  }
}

// ---------------------------------------------------------------------------
extern "C" void kernel_launch(void* const* d_in, const int* in_sizes, int n_in,
                              void* d_out, int out_size, void* d_ws, size_t ws_size,
                              hipStream_t stream) {
  (void)in_sizes; (void)n_in; (void)out_size; (void)ws_size;
  const float* x    = (const float*)d_in[0];
  const float* hi   = (const float*)d_in[1];
  const float* Wih  = (const float*)d_in[2];
  const float* Whh  = (const float*)d_in[3];
  const float* bih  = (const float*)d_in[4];
  const float* bhh  = (const float*)d_in[5];
  float* out = (float*)d_out;

  unsigned* bar = (unsigned*)((char*)d_ws + kBarOff);
  float* hbuf   = (float*)((char*)d_ws + kHbufOff);
  float* Gx     = (float*)((char*)d_ws + kGxOff);   // needs ~134 MB of d_ws

  // Zero barrier state + h ping-pong each call (capture-safe).
  hipMemsetAsync(d_ws, 0, kGxOff, stream);

  // Phase A: 128 (M) x 32 (N) blocks of 256 threads.
  pregate_gemm<<<dim3(128 * 32), dim3(256), 0, stream>>>(x, hi, Wih, bih, bhh, Gx);

  // Phase B: 16 cooperative WGs, 308 KB dynamic LDS each (W_hh slice resident).
  lstm_scan<<<dim3(kNWG), dim3(256), kSmemBytes, stream>>>(Whh, Wih, Gx, out, hbuf, bar);
}